// LearnedEquivariance1D_57114475102851
// MI455X (gfx1250) — compile-verified
//
#include <hip/hip_runtime.h>

// CDNA5 (gfx1250) wave32 WMMA types
typedef __attribute__((ext_vector_type(2))) float v2f;
typedef __attribute__((ext_vector_type(8))) float v8f;

#define L_LEN    65536
#define B_ROWS   128
#define KW       65
#define HALF     32
#define GS       128
#define CHUNK    8192
#define KPAD     84               // 21 * 4 (K padded for 16x16x4 WMMA steps)
#define KSTEPS   (KPAD / 4)       // 21
#define NTHREADS 256
#define NWAVES   8
#define TILES_PER_WAVE 4          // CHUNK / 256 / NWAVES

// One workgroup per row. Whole row lives in LDS (256KB of the 320KB/WGP),
// all 8 conv iterations run out of LDS; matrix math on V_WMMA_F32_16X16X4_F32.
// The banded Toeplitz B-matrix is preloaded into 42 VGPRs and reused by every
// WMMA in the kernel.
__global__ __launch_bounds__(NTHREADS, 1)
void conv_rows_wmma_kernel(const float* __restrict__ x,
                           const long long* __restrict__ g,
                           const float* __restrict__ kernels,
                           const int* __restrict__ n_ptr,
                           float* __restrict__ out)
{
    __shared__ float rowbuf[L_LEN];            // 262144 B : full row, iterated in place
    __shared__ float stage[CHUNK + 96];        //  33152 B : old-value window (+halos, +zero pad)
    __shared__ float Bs[KPAD * 16];            //   5376 B : banded Toeplitz of the 65-tap kernel
    __shared__ float halo_prev[HALF];          // old x[c-32..c) carried between chunks
    __shared__ float halo_head[HALF];          // old x[0..32) saved for circular wrap

    const int tid  = threadIdx.x;
    const int lane = tid & 31;
    const int wave = tid >> 5;
    const int b    = blockIdx.x;

    const int niter = n_ptr[0];

    // ---- build Bs[j][n] = w[j-n] (0 <= j-n < 65), else 0 -------------------
    {
        long long gv = g[b];                        // g dtype is int64 in reference
        int kidx = (int)(gv & (long long)(GS - 1)); // g % 128, g >= 0
        const float* w = kernels + kidx * KW;
        for (int idx = tid; idx < KPAD * 16; idx += NTHREADS) {
            int j = idx >> 4;
            int n = idx & 15;
            int d = j - n;
            Bs[idx] = (d >= 0 && d < KW) ? w[d] : 0.0f;
        }
    }

    // ---- load the row once from HBM into LDS (float4 vectorized) ----------
    {
        const float4* src = (const float4*)(x + (size_t)b * L_LEN);
        float4* dst = (float4*)rowbuf;
        for (int i = tid; i < L_LEN / 4; i += NTHREADS) dst[i] = src[i];
    }
    __syncthreads();

    // per-lane WMMA fragment geometry (wave32):
    // A 16x4 f32: lanes 0-15 hold (K=0,K=1), lanes 16-31 hold (K=2,K=3)
    // B 4x16 f32: VGPR0 = rows K0(l0-15)/K2(l16-31), VGPR1 = K1/K3
    const int m_lo  = (lane < 16) ? lane : (lane - 16);
    const int kbase = (lane < 16) ? 0 : 2;
    const int n_col = m_lo;

    // ---- hoist all 21 B fragments into registers (live for whole kernel) --
    v2f bfr[KSTEPS];
    #pragma unroll
    for (int jt = 0; jt < KSTEPS; ++jt) {
        const int j0 = 4 * jt + kbase;
        bfr[jt].x = Bs[j0 * 16 + n_col];
        bfr[jt].y = Bs[(j0 + 1) * 16 + n_col];
    }

    for (int it = 0; it < niter; ++it) {
        // snapshot circular-wrap halos of the *old* row
        if (tid < HALF) {
            halo_head[tid] = rowbuf[tid];
            halo_prev[tid] = rowbuf[L_LEN - HALF + tid];
        }
        __syncthreads();

        for (int c = 0; c < L_LEN; c += CHUNK) {
            // ---- stage[t] = old x[c - 32 + t], t in [0, CHUNK+64); zero tail
            for (int t = tid; t < CHUNK + 96; t += NTHREADS) {
                float v;
                if (t < HALF) {
                    v = halo_prev[t];
                } else if (t < CHUNK + HALF) {
                    v = rowbuf[c + t - HALF];
                } else if (t < CHUNK + 2 * HALF) {
                    int d = t - (CHUNK + HALF);
                    int idx2 = c + CHUNK + d;
                    v = (idx2 < L_LEN) ? rowbuf[idx2] : halo_head[d]; // wrap on last chunk
                } else {
                    v = 0.0f;   // pad for K over-read (matching B rows are all zero)
                }
                stage[t] = v;
            }
            __syncthreads();

            // ---- each wave computes 4 16x16 output tiles via 21 fp32 WMMAs each
            v8f acc[TILES_PER_WAVE];
            #pragma unroll
            for (int tt = 0; tt < TILES_PER_WAVE; ++tt) acc[tt] = (v8f){};

            #pragma unroll
            for (int jt = 0; jt < KSTEPS; ++jt) {
                const int j0 = 4 * jt + kbase;
                const v2f bv = bfr[jt];
                #pragma unroll
                for (int tt = 0; tt < TILES_PER_WAVE; ++tt) {
                    const int toff = 256 * (wave + NWAVES * tt);
                    const v2f a = *(const v2f*)&stage[toff + 16 * m_lo + j0];
                    acc[tt] = __builtin_amdgcn_wmma_f32_16x16x4_f32(
                                  false, a, false, bv, (short)0, acc[tt], false, false);
                }
            }

            // carry old-value halo for next chunk (still old in stage)
            if (tid < HALF) halo_prev[tid] = stage[CHUNK + tid];

            // ---- scatter C/D fragments back into rowbuf (in-place update)
            // VGPR r: lanes 0-15 -> M=r, N=lane; lanes 16-31 -> M=r+8, N=lane-16
            const int mrow = (lane < 16) ? 0 : 8;
            #pragma unroll
            for (int tt = 0; tt < TILES_PER_WAVE; ++tt) {
                const int p0 = c + 256 * (wave + NWAVES * tt);
                #pragma unroll
                for (int r = 0; r < 8; ++r) {
                    rowbuf[p0 + 16 * (r + mrow) + n_col] = acc[tt][r];
                }
            }
            __syncthreads();
        }
    }

    // ---- write the finished row back to HBM once ---------------------------
    {
        const float4* src = (const float4*)rowbuf;
        float4* dst = (float4*)(out + (size_t)b * L_LEN);
        for (int i = tid; i < L_LEN / 4; i += NTHREADS) dst[i] = src[i];
    }
}

extern "C" void kernel_launch(void* const* d_in, const int* in_sizes, int n_in,
                              void* d_out, int out_size, void* d_ws, size_t ws_size,
                              hipStream_t stream) {
    (void)in_sizes; (void)n_in; (void)d_ws; (void)ws_size; (void)out_size;
    const float*     x       = (const float*)d_in[0];
    const long long* g       = (const long long*)d_in[1];  // int64 per reference
    const float*     kernels = (const float*)d_in[2];
    // d_in[3] = l (ignored: num_layers == 0 in reference)
    const int*       n_ptr   = (const int*)d_in[4];
    float*           out     = (float*)d_out;

    conv_rows_wmma_kernel<<<dim3(B_ROWS), dim3(NTHREADS), 0, stream>>>(
        x, g, kernels, n_ptr, out);
}